// GCN_29119878267593
// MI455X (gfx1250) — compile-verified
//
#include <hip/hip_runtime.h>

typedef __attribute__((ext_vector_type(2))) float v2f;
typedef __attribute__((ext_vector_type(8))) float v8f;

// ---------------------------------------------------------------------------
// GEMM  C[M,N] = A[M,K] @ W[K,N]   (fp32, V_WMMA_F32_16X16X4_F32)
// One wave per 16-row tile; all N/16 column tiles accumulated per wave so A is
// read exactly once. W staged in LDS pre-paired as float2 (K, K+1) so each
// lane's B fragment is a single 8B LDS read; +8 float2 row pad de-conflicts
// the two 16-lane halves. Requires M % 16 == 0 (true here: 100000).
// A-frag  (16x4 f32): lane&15 = row, lane>>4 selects K pair {0,1} vs {2,3}.
// B-frag  (4x16 f32): lane&15 = col, lane>>4 selects K pair.
// C/D     (16x16 f32): VGPR j holds rows j (lanes 0-15) and j+8 (lanes 16-31).
// ---------------------------------------------------------------------------
template <int K, int N>
__global__ __launch_bounds__(256) void gemm_wmma_f32(
    const float* __restrict__ A, const float* __restrict__ W,
    float* __restrict__ C, int M) {
  constexpr int LDW = N + 8;  // padded stride in float2 units
  __shared__ v2f lW[(K / 2) * LDW];
  for (int idx = threadIdx.x; idx < (K / 2) * N; idx += 256) {
    int kk = idx / N, col = idx % N;
    v2f t;
    t[0] = W[(2 * kk) * N + col];
    t[1] = W[(2 * kk + 1) * N + col];
    lW[kk * LDW + col] = t;
  }
  __syncthreads();

  const int lane = threadIdx.x & 31;
  const int wv   = threadIdx.x >> 5;
  const int tile = blockIdx.x * 8 + wv;
  if (tile * 16 >= M) return;  // wave-uniform: EXEC stays all-ones for WMMA

  const int r    = lane & 15;
  const int half = lane >> 4;
  const float* Arow = A + (size_t)(tile * 16 + r) * K;

  v8f acc[N / 16];
#pragma unroll
  for (int i = 0; i < N / 16; ++i) acc[i] = (v8f){0.f, 0.f, 0.f, 0.f, 0.f, 0.f, 0.f, 0.f};

#pragma unroll
  for (int k = 0; k < K; k += 4) {
    const v2f a = *(const v2f*)(Arow + k + 2 * half);  // 8B global load
#pragma unroll
    for (int nt = 0; nt < N / 16; ++nt) {
      const v2f b = lW[(k / 2 + half) * LDW + nt * 16 + r];  // 8B LDS load
      acc[nt] = __builtin_amdgcn_wmma_f32_16x16x4_f32(
          /*neg_a=*/false, a, /*neg_b=*/false, b,
          /*c_mod=*/(short)0, acc[nt], /*reuse_a=*/false, /*reuse_b=*/false);
    }
  }

#pragma unroll
  for (int nt = 0; nt < N / 16; ++nt) {
    float* Cb = C + (size_t)(tile * 16) * N + nt * 16 + r;
#pragma unroll
    for (int j = 0; j < 8; ++j) Cb[(size_t)(j + half * 8) * N] = acc[nt][j];
  }
}

// ---------------------------------------------------------------------------
// Degree / normalization helpers
// ---------------------------------------------------------------------------
__global__ void k_fill1(float* __restrict__ p, int n) {
  int i = blockIdx.x * 256 + threadIdx.x;
  if (i < n) p[i] = 1.0f;  // self-loop contributes 1 to every in-degree
}

__global__ void k_degacc(const long long* __restrict__ dst, float* __restrict__ deg, int E) {
  int i = blockIdx.x * 256 + threadIdx.x;
  if (i < E) atomicAdd(&deg[(size_t)dst[i]], 1.0f);
}

__global__ void k_rsqrt(float* __restrict__ p, int n) {
  int i = blockIdx.x * 256 + threadIdx.x;
  if (i < n) p[i] = rsqrtf(p[i]);  // v_rsq_f32
}

// out[i,:] = h[i,:] * dinv[i]^2 + bias   (self-loop term + bias, fused init)
template <int F>
__global__ void k_self_init(const float* __restrict__ h, const float* __restrict__ dinv,
                            const float* __restrict__ bias, float* __restrict__ out, int n) {
  constexpr int C4 = F / 4;
  long long t = (long long)blockIdx.x * 256 + threadIdx.x;
  int i = (int)(t / C4), c = (int)(t % C4);
  if (i >= n) return;
  float di = dinv[i];
  float s = di * di;
  float4 hv = ((const float4*)(h + (size_t)i * F))[c];
  float4 bv = ((const float4*)bias)[c];
  float4 o = make_float4(hv.x * s + bv.x, hv.y * s + bv.y,
                         hv.z * s + bv.z, hv.w * s + bv.w);
  ((float4*)(out + (size_t)i * F))[c] = o;
}

// out[dst,:] += h[src,:] * dinv[src]*dinv[dst]
// 16 (or 8) consecutive lanes cover one edge -> 256B (or 128B) coalesced gather of h[src].
template <int F>
__global__ void k_scatter(const long long* __restrict__ src, const long long* __restrict__ dst,
                          const float* __restrict__ h, const float* __restrict__ dinv,
                          float* __restrict__ out, int E) {
  constexpr int C4 = F / 4;
  long long t = (long long)blockIdx.x * 256 + threadIdx.x;
  int e = (int)(t / C4), c = (int)(t % C4);
  if (e >= E) return;
  size_t s = (size_t)src[e], d = (size_t)dst[e];
  float nrm = dinv[s] * dinv[d];
  float4 m = ((const float4*)(h + s * F))[c];
  float* o = out + d * F + (size_t)c * 4;
  atomicAdd(o + 0, m.x * nrm);  // global_atomic_add_f32
  atomicAdd(o + 1, m.y * nrm);
  atomicAdd(o + 2, m.z * nrm);
  atomicAdd(o + 3, m.w * nrm);
}

__global__ void k_relu4(float4* __restrict__ p, long long n4) {
  long long i = (long long)blockIdx.x * 256 + threadIdx.x;
  if (i < n4) {
    float4 v = p[i];
    v.x = fmaxf(v.x, 0.f); v.y = fmaxf(v.y, 0.f);
    v.z = fmaxf(v.z, 0.f); v.w = fmaxf(v.w, 0.f);
    p[i] = v;
  }
}

// ---------------------------------------------------------------------------
// Launch: two GCN layers. inputs: x[N,128], edge_index[2,E] (i64),
// W1[128,64], b1[64], W2[64,32], b2[32]; output: [N,32] f32.
// ---------------------------------------------------------------------------
extern "C" void kernel_launch(void* const* d_in, const int* in_sizes, int n_in,
                              void* d_out, int out_size, void* d_ws, size_t ws_size,
                              hipStream_t stream) {
  const float*     x   = (const float*)d_in[0];
  const long long* ei  = (const long long*)d_in[1];
  const float*     W1  = (const float*)d_in[2];
  const float*     b1  = (const float*)d_in[3];
  const float*     W2  = (const float*)d_in[4];
  const float*     b2  = (const float*)d_in[5];
  float*           out = (float*)d_out;

  const int n = in_sizes[0] / 128;  // 100000 (multiple of 16)
  const int E = in_sizes[1] / 2;    // 1600000
  const long long* src = ei;
  const long long* dst = ei + E;

  // Workspace layout (~64.5 MB): dinv | h[n,64] | agg1[n,64] | h2[n,32]
  char* ws = (char*)d_ws;
  size_t off = ((size_t)n * 4 + 255) & ~(size_t)255;
  float* dinv = (float*)ws;
  float* h    = (float*)(ws + off);
  float* agg1 = h + (size_t)n * 64;
  float* h2   = agg1 + (size_t)n * 64;

  auto nb = [](long long work) { return (unsigned)((work + 255) / 256); };

  // normalization: deg (incl. self loop) -> dinv = rsqrt(deg)
  k_fill1<<<nb(n), 256, 0, stream>>>(dinv, n);
  k_degacc<<<nb(E), 256, 0, stream>>>(dst, dinv, E);
  k_rsqrt<<<nb(n), 256, 0, stream>>>(dinv, n);

  const unsigned gemm_blocks = (unsigned)((n / 16 + 7) / 8);

  // layer 1: h = x@W1 ; agg1 = self + edges ; relu (bias fused into init)
  gemm_wmma_f32<128, 64><<<gemm_blocks, 256, 0, stream>>>(x, W1, h, n);
  k_self_init<64><<<nb((long long)n * 16), 256, 0, stream>>>(h, dinv, b1, agg1, n);
  k_scatter<64><<<nb((long long)E * 16), 256, 0, stream>>>(src, dst, h, dinv, agg1, E);
  k_relu4<<<nb((long long)n * 16), 256, 0, stream>>>((float4*)agg1, (long long)n * 16);

  // layer 2: h2 = agg1@W2 ; out = self + b2 + edges
  gemm_wmma_f32<64, 32><<<gemm_blocks, 256, 0, stream>>>(agg1, W2, h2, n);
  k_self_init<32><<<nb((long long)n * 8), 256, 0, stream>>>(h2, dinv, b2, out, n);
  k_scatter<32><<<nb((long long)E * 8), 256, 0, stream>>>(src, dst, h2, dinv, out, E);
}